// hyper_graph_conv_net_53137335386354
// MI455X (gfx1250) — compile-verified
//
#include <hip/hip_runtime.h>

// Hyper-graph conv: 2 layers of bipartite SpMM (COO, atomic scatter) + mean + L2 row-normalize.
// SpMM is memory/atomic bound (L2-resident on MI455X's 192MB L2) -> wave-per-edge float4 gather,
// global_atomic_add_f32 scatter. Final row-norm uses V_WMMA_F32_16X16X4_F32 with an all-ones B
// matrix as a zero-shuffle cross-lane sum-of-squares reduction (16 rows per wave).

typedef __attribute__((ext_vector_type(2))) float v2f;
typedef __attribute__((ext_vector_type(8))) float v8f;

#define DIM      128
#define NBUNDLES 50000
#define NLAYER   2

__global__ __launch_bounds__(256) void init_copy2(const float4* __restrict__ src,
                                                  float4* __restrict__ a,
                                                  float4* __restrict__ b, int n4) {
  int i = blockIdx.x * blockDim.x + threadIdx.x;
  if (i < n4) { float4 v = src[i]; a[i] = v; b[i] = v; }
}

__global__ __launch_bounds__(256) void zero4(float4* __restrict__ p, int n4) {
  int i = blockIdx.x * blockDim.x + threadIdx.x;
  if (i < n4) p[i] = make_float4(0.f, 0.f, 0.f, 0.f);
}

__global__ __launch_bounds__(256) void addacc4(float4* __restrict__ acc,
                                               const float4* __restrict__ x, int n4) {
  int i = blockIdx.x * blockDim.x + threadIdx.x;
  if (i < n4) {
    float4 a = acc[i]; float4 v = x[i];
    a.x += v.x; a.y += v.y; a.z += v.z; a.w += v.w;
    acc[i] = a;
  }
}

// One wave32 per edge: lane l moves floats [4l..4l+3] of the 128-float row.
// Gather is one coalesced 512B row read; scatter is 4 f32 atomics/lane into L2.
__global__ __launch_bounds__(256) void spmm_atomic(const int* __restrict__ dst_rows,
                                                   const int* __restrict__ src_cols,
                                                   const float* __restrict__ vals,
                                                   const float* __restrict__ x,
                                                   float* __restrict__ y, int nnz) {
  int e    = (int)((blockIdx.x * blockDim.x + threadIdx.x) >> 5);
  int lane = threadIdx.x & 31;
  if (e >= nnz) return;                       // wave-uniform guard
  int   r = dst_rows[e];
  int   c = src_cols[e];
  float v = vals[e];
  float4 d = *(const float4*)(x + (size_t)c * DIM + lane * 4);
  float* dst = y + (size_t)r * DIM + lane * 4;
  atomicAdd(dst + 0, v * d.x);
  atomicAdd(dst + 1, v * d.y);
  atomicAdd(dst + 2, v * d.z);
  atomicAdd(dst + 3, v * d.w);
}

// Per wave: 16 rows. WMMA f32 16x16x4 with B = ones computes D[m,n] = sum_k A[m,k],
// feeding A = acc^2 gives each lane 8 of the 16 row sums-of-squares (replicated over N).
// out = acc / max(sqrt(S), (num_layer+1)*1e-12)  ==  (acc/3) / max(||acc/3||, 1e-12).
__global__ __launch_bounds__(256) void mean_l2norm_wmma(float* __restrict__ out, int nrows) {
  int gwave = (int)((blockIdx.x * blockDim.x + threadIdx.x) >> 5);
  int lane  = threadIdx.x & 31;
  int row0  = gwave * 16;
  if (row0 >= nrows) return;                  // wave-uniform: EXEC all-ones below
  float* base = out + (size_t)row0 * DIM;

  const int m  = lane & 15;                   // A-matrix: lane -> row M
  const int kh = (lane >> 4) * 2;             // lanes 16-31 hold K+2,K+3 halves

  v8f c = {};
  v2f bones; bones.x = 1.0f; bones.y = 1.0f;  // B = all ones (layout-invariant)

#pragma unroll
  for (int k0 = 0; k0 < DIM; k0 += 4) {
    float2 av = *(const float2*)(base + (size_t)m * DIM + k0 + kh);
    v2f a; a.x = av.x * av.x; a.y = av.y * av.y;
    c = __builtin_amdgcn_wmma_f32_16x16x4_f32(
        /*neg_a=*/false, a, /*neg_b=*/false, bones,
        /*c_mod=*/(short)0, c, /*reuse_a=*/false, /*reuse_b=*/false);
  }

  // D layout: VGPR j, lanes 0-15 -> M=j ; lanes 16-31 -> M=8+j. Value identical over N.
  const int rsel    = (lane >> 4) * 8;
  const int colbase = (lane & 15) * 8;        // each lane rescales 8 rows x 8 cols (disjoint)
#pragma unroll
  for (int j = 0; j < 8; ++j) {
    float s   = c[j];
    float inv = 1.0f / fmaxf(sqrtf(s), 3e-12f);
    float* p  = base + (size_t)(rsel + j) * DIM + colbase;
    float4 a0 = *(float4*)(p);
    float4 a1 = *(float4*)(p + 4);
    a0.x *= inv; a0.y *= inv; a0.z *= inv; a0.w *= inv;
    a1.x *= inv; a1.y *= inv; a1.z *= inv; a1.w *= inv;
    *(float4*)(p)     = a0;
    *(float4*)(p + 4) = a1;
  }
}

extern "C" void kernel_launch(void* const* d_in, const int* in_sizes, int n_in,
                              void* d_out, int out_size, void* d_ws, size_t ws_size,
                              hipStream_t stream) {
  const float* item_emb = (const float*)d_in[0];
  const float* bi_vals  = (const float*)d_in[1];
  const float* ib_vals  = (const float*)d_in[2];
  const int*   bi_rows  = (const int*)d_in[3];
  const int*   bi_cols  = (const int*)d_in[4];

  const int nnz       = in_sizes[1];
  const int num_items = in_sizes[0] / DIM;

  float* out   = (float*)d_out;                       // doubles as the accumulator
  float* feats = (float*)d_ws;                        // num_items * 128 f32 (102.4 MB)
  float* msg   = feats + (size_t)num_items * DIM;     // NBUNDLES * 128 f32 ( 25.6 MB)

  const int TB       = 256;
  const int n4_items = num_items * DIM / 4;
  const int n4_bund  = NBUNDLES * DIM / 4;

  // feats = item_emb ; acc(out) = item_emb
  init_copy2<<<(n4_items + TB - 1) / TB, TB, 0, stream>>>(
      (const float4*)item_emb, (float4*)feats, (float4*)out, n4_items);

  const int spmm_blocks = (int)(((long long)nnz * 32 + TB - 1) / TB);

  for (int l = 0; l < NLAYER; ++l) {
    // hop 1: bundles aggregate items
    zero4<<<(n4_bund + TB - 1) / TB, TB, 0, stream>>>((float4*)msg, n4_bund);
    spmm_atomic<<<spmm_blocks, TB, 0, stream>>>(bi_rows, bi_cols, bi_vals, feats, msg, nnz);
    // hop 2: items aggregate bundles
    zero4<<<(n4_items + TB - 1) / TB, TB, 0, stream>>>((float4*)feats, n4_items);
    spmm_atomic<<<spmm_blocks, TB, 0, stream>>>(bi_cols, bi_rows, ib_vals, msg, feats, nnz);
    // acc += feats
    addacc4<<<(n4_items + TB - 1) / TB, TB, 0, stream>>>((float4*)out, (const float4*)feats,
                                                         n4_items);
  }

  // mean + L2 normalize in place (WMMA row-reduction, 16 rows per wave)
  const int nwaves = (num_items + 15) / 16;
  mean_l2norm_wmma<<<(nwaves * 32 + TB - 1) / TB, TB, 0, stream>>>(out, num_items);
}